// MVDR_32985348834005
// MI455X (gfx1250) — compile-verified
//
#include <hip/hip_runtime.h>

// MVDR beamformer for MI455X (gfx1250, wave32).
// B=4, C=8, F=129, T=800. Complex 8x8 Hermitian algebra embedded as
// symmetric 16x16 real matrices executed on V_WMMA_F32_16X16X4_F32.
//
// Key structure:
//  * kernel 1: chunk-boundary PSD prefix sums (2 timesteps per WMMA).
//  * kernel 2: per (b,f,chunk) wave walks t sequentially: WMMA PSD update,
//    Tikhonov loading folded into the unnormalized accumulator, warm-started
//    Newton-Schulz inverse (pure 16x16 WMMA matmuls), numerator = Minv@psd_s,
//    trace normalization, w^H x application.
//  * Symmetric matrices (psd_s, loaded psd_n, Minv) use a single LDS-loaded
//    operand image for both A- and B-layout (images coincide for M == M^T).

#define Bn 4
#define Cn 8
#define Fn 129
#define Tn 800
#define F2 (2 * Fn)
#define DIAG_EPS 1e-7f
#define EPSV 1e-8f
#define NCHUNK 4
#define TC (Tn / NCHUNK)       // 200
#define LDSTR 17               // padded LDS row stride (dwords) for 16x16 tile

typedef __attribute__((ext_vector_type(2))) float v2f;
typedef __attribute__((ext_vector_type(8))) float v8f;

static __device__ inline v8f zero8() {
  v8f z = {0.f, 0.f, 0.f, 0.f, 0.f, 0.f, 0.f, 0.f};
  return z;
}

// One 16x16 += 16x4 @ 4x16 step on the f32 matrix pipe.
static __device__ inline v8f wmma4(float a0, float a1, float b0, float b1, v8f c) {
  v2f a = {a0, a1};
  v2f b = {b0, b1};
  // 8 args: (neg_a, A, neg_b, B, c_mod, C, reuse_a, reuse_b)
  return __builtin_amdgcn_wmma_f32_16x16x4_f32(false, a, false, b, (short)0, c,
                                               false, false);
}

// Full 16x16 @ 16x16 (K accumulated over 4 chunks of 4).
static __device__ inline v8f mm16(const float* a, const float* b, v8f c) {
#pragma unroll
  for (int kc = 0; kc < 4; ++kc)
    c = wmma4(a[2 * kc], a[2 * kc + 1], b[2 * kc], b[2 * kc + 1], c);
  return c;
}

// C/D layout: VGPR r of lane l holds M = r + 8*(l>>4), N = l&15.
static __device__ inline void store_c(float* buf, v8f c, int n, int hi) {
#pragma unroll
  for (int r = 0; r < 8; ++r) buf[(r + 8 * hi) * LDSTR + n] = c[r];
}

// A layout (left operand), chunk kc: VGPR0 = A[m][4kc + (hi?2:0)], VGPR1 = +1.
// For SYMMETRIC matrices this register image is also a valid B-layout image.
static __device__ inline void load_a(const float* buf, float* a, int m, int hi) {
#pragma unroll
  for (int kc = 0; kc < 4; ++kc) {
    a[2 * kc]     = buf[m * LDSTR + 4 * kc + (hi ? 2 : 0)];
    a[2 * kc + 1] = buf[m * LDSTR + 4 * kc + (hi ? 3 : 1)];
  }
}

// B layout (right operand), chunk kc: VGPR0 = B[4kc + (hi?2:0)][n], VGPR1 = +1.
static __device__ inline void load_b(const float* buf, float* bq, int n, int hi) {
#pragma unroll
  for (int kc = 0; kc < 4; ++kc) {
    bq[2 * kc]     = buf[(4 * kc + (hi ? 2 : 0)) * LDSTR + n];
    bq[2 * kc + 1] = buf[(4 * kc + (hi ? 3 : 1)) * LDSTR + n];
  }
}

// Trace of the 16x16 real embedding (= 2 * Re(trace) of the complex 8x8).
static __device__ inline float tr16(v8f c, bool hasDiag, int rdiag) {
  float d = 0.f;
#pragma unroll
  for (int r = 0; r < 8; ++r) d = (hasDiag && (r == rdiag)) ? c[r] : d;
#pragma unroll
  for (int off = 16; off > 0; off >>= 1) d += __shfl_xor(d, off, 32);
  return d;
}

// ---------------------------------------------------------------------------
// Kernel 1: chunk-boundary PSD prefix sums. One wave per (b,f); WMMA
// accumulates two timesteps per instruction (K=4 = 2 x [re|im] columns).
// ---------------------------------------------------------------------------
__global__ __launch_bounds__(32) void mvdr_psd_bases(const float* __restrict__ X,
                                                     const float* __restrict__ S,
                                                     float* __restrict__ ws) {
  const int bf = blockIdx.x;                 // 0 .. B*F-1
  const int b = bf / Fn, f = bf % Fn;
  const int lane = (int)threadIdx.x;
  const int m = lane & 15, hi = lane >> 4;   // row of augmented matrix, t-half
  const int ch = m & 7;
  const bool top = (m < 8);                  // real-part row vs imag-part row
  const size_t base_re = (((size_t)b * Cn + ch) * F2 + f) * Tn;
  const size_t base_im = (((size_t)b * Cn + ch) * F2 + f + Fn) * Tn;

  v8f accS = zero8();
  v8f accN = zero8();

  for (int t = 0; t < TC * (NCHUNK - 1); t += 2) {
    const int tt = t + hi;                   // lanes 0-15: t, lanes 16-31: t+1
    const float sr = S[base_re + tt], si = S[base_im + tt];
    const float xr = X[base_re + tt], xi = X[base_im + tt];
    const float nr = xr - sr, ni = xi - si;
    // Z2 = [[zr, -zi],[zi, zr]] -> A operand == B operand register image.
    const float as0 = top ? sr : si, as1 = top ? -si : sr;
    accS = wmma4(as0, as1, as0, as1, accS);
    const float an0 = top ? nr : ni, an1 = top ? -ni : nr;
    accN = wmma4(an0, an1, an0, an1, accN);

    if (((t + 2) % TC) == 0) {               // snapshot exclusive prefix
      const int k = (t + 2) / TC;            // 1..NCHUNK-1
      float* slotS = ws + ((((size_t)bf * (NCHUNK - 1)) + (k - 1)) * 2) * 256;
      float* slotN = slotS + 256;
#pragma unroll
      for (int r = 0; r < 8; ++r) {
        slotS[(r + 8 * hi) * 16 + m] = accS[r];
        slotN[(r + 8 * hi) * 16 + m] = accN[r];
      }
    }
  }
}

// ---------------------------------------------------------------------------
// Kernel 2: per (b,f,chunk) wave.
// ---------------------------------------------------------------------------
__global__ __launch_bounds__(32) void mvdr_solve_apply(const float* __restrict__ X,
                                                       const float* __restrict__ S,
                                                       const float* __restrict__ ws,
                                                       float* __restrict__ out) {
  __shared__ float ldsA[16 * LDSTR];   // loaded noise PSD A'
  __shared__ float ldsS[16 * LDSTR];   // source PSD
  __shared__ float ldsT1[16 * LDSTR];  // temp: Minv round-trips
  __shared__ float ldsT2[16 * LDSTR];  // temp: R = 2I - A'Minv round-trips

  const int bid = blockIdx.x;
  const int k  = bid % NCHUNK;
  const int bf = bid / NCHUNK;
  const int b = bf / Fn, f = bf % Fn;
  const int lane = (int)threadIdx.x;
  const int m = lane & 15, hi = lane >> 4;
  const int ch = m & 7;
  const bool top = (m < 8);
  const bool lo16 = (lane < 16);
  const int rdiag = m - 8 * hi;                       // which VGPR holds diag
  const bool hasDiag = (rdiag >= 0) && (rdiag < 8);

  const size_t base_re = (((size_t)b * Cn + ch) * F2 + f) * Tn;
  const size_t base_im = (((size_t)b * Cn + ch) * F2 + f + Fn) * Tn;
  const int t0 = k * TC;

  // Warm L2 for this wave's sequential t-walk (global_prefetch_b8).
  __builtin_prefetch(&S[base_re + t0], 0, 1);
  __builtin_prefetch(&S[base_im + t0], 0, 1);
  __builtin_prefetch(&X[base_re + t0], 0, 1);
  __builtin_prefetch(&X[base_im + t0], 0, 1);

  v8f accS = zero8();
  v8f accN = zero8();
  if (k > 0) {                                        // chunk prefix base
    const float* slotS = ws + ((((size_t)bf * (NCHUNK - 1)) + (k - 1)) * 2) * 256;
    const float* slotN = slotS + 256;
#pragma unroll
    for (int r = 0; r < 8; ++r) {
      accS[r] = slotS[(r + 8 * hi) * 16 + m];
      accN[r] = slotN[(r + 8 * hi) * 16 + m];
    }
  }

  v8f minv = zero8();

  for (int t = t0; t < t0 + TC; ++t) {
    // ---- load this timestep, update PSDs (single-t: zero K=2,3 columns) ----
    const float sr = S[base_re + t], si = S[base_im + t];
    const float xr = X[base_re + t], xi = X[base_im + t];
    const float nr = xr - sr, ni = xi - si;
    const float as0 = lo16 ? (top ? sr : si) : 0.f;
    const float as1 = lo16 ? (top ? -si : sr) : 0.f;
    accS = wmma4(as0, as1, as0, as1, accS);
    const float an0 = lo16 ? (top ? nr : ni) : 0.f;
    const float an1 = lo16 ? (top ? -ni : nr) : 0.f;
    accN = wmma4(an0, an1, an0, an1, accN);

    // ---- Tikhonov loading on the UNNORMALIZED accumulator (1/(t+1) cancels)
    const float trN = tr16(accN, hasDiag, rdiag);     // trace of real embed
    const float loadv = 0.5f * trN * DIAG_EPS + EPSV * (float)(t + 1);
    v8f aP = accN;
#pragma unroll
    for (int r = 0; r < 8; ++r)
      aP[r] += (hasDiag && (r == rdiag)) ? loadv : 0.f;

    __syncthreads();                // WAR: previous t's aA/bS loads complete
    store_c(ldsA, aP, m, hi);
    store_c(ldsS, accS, m, hi);
    __syncthreads();
    float aA[8], bS[8];
    load_a(ldsA, aA, m, hi);        // A' as left operand (symmetric image)
    load_b(ldsS, bS, m, hi);        // psd_s as right operand

    // ---- Newton-Schulz: minv <- minv (2I - A' minv) ----
    if (t == t0) {                  // cold start: X0 = I/tr(A')
      const float c0 = 1.f / (0.5f * trN + 8.f * loadv + 1e-30f);
#pragma unroll
      for (int r = 0; r < 8; ++r)
        minv[r] = (hasDiag && (r == rdiag)) ? c0 : 0.f;
    }
    const int iters = (t == t0) ? 30 : 2;             // warm start after t0
    for (int it = 0; it < iters; ++it) {
      __syncthreads();              // WAR on T1 (prev M loads) + T2 (prev bR)
      store_c(ldsT1, minv, m, hi);
      __syncthreads();
      float Mim[8];                 // Minv symmetric: one image = A and B op
      load_a(ldsT1, Mim, m, hi);
      v8f t1 = mm16(aA, Mim, zero8());                // A' @ Minv
      v8f rr;
#pragma unroll
      for (int r = 0; r < 8; ++r)
        rr[r] = ((hasDiag && (r == rdiag)) ? 2.f : 0.f) - t1[r];
      store_c(ldsT2, rr, m, hi);    // R not symmetric: needs true B image
      __syncthreads();
      float bR[8];
      load_b(ldsT2, bR, m, hi);
      minv = mm16(Mim, bR, zero8());                  // Minv @ (2I - A'Minv)
    }

    // ---- numerator = Minv @ psd_s ; ws = num / (trace(num)+eps) ----
    __syncthreads();
    store_c(ldsT1, minv, m, hi);
    __syncthreads();
    float M2[8];
    load_a(ldsT1, M2, m, hi);       // symmetric image, used as left operand
    v8f num = mm16(M2, bS, zero8());
    const float trNum = 0.5f * tr16(num, hasDiag, rdiag);  // Re(complex trace)
    const float invden = 1.f / (trNum + EPSV);

    // ---- enh = ws^H x  (real embed: ws16^T @ [xr; xi]) ----
    float acc = 0.f;
#pragma unroll
    for (int r = 0; r < 8; ++r) {
      const float vlo = __shfl(xr, r, 32);            // xv[r]   = Re x_r
      const float vhi = __shfl(xi, r, 32);            // xv[r+8] = Im x_r
      acc += num[r] * (lo16 ? vlo : vhi);
    }
    acc += __shfl_xor(acc, 16, 32);                   // combine row halves
    const float val = acc * invden;
    // Scattered 4B stores are fine: full output (26 MB) stays resident in the
    // 192 MB L2, so write-back to HBM coalesces at cache-line granularity.
    if (lo16) {
      const size_t o =
          (((size_t)b * Cn + ch) * F2 + (top ? f : f + Fn)) * (size_t)Tn + t;
      out[o] = val;
    }
  }
}

extern "C" void kernel_launch(void* const* d_in, const int* in_sizes, int n_in,
                              void* d_out, int out_size, void* d_ws, size_t ws_size,
                              hipStream_t stream) {
  (void)in_sizes; (void)n_in; (void)out_size; (void)ws_size;
  const float* X = (const float*)d_in[0];
  const float* S = (const float*)d_in[1];
  float* out = (float*)d_out;
  float* wsf = (float*)d_ws;   // B*F*(NCHUNK-1)*2*256 floats ≈ 3.2 MB

  mvdr_psd_bases<<<Bn * Fn, 32, 0, stream>>>(X, S, wsf);
  mvdr_solve_apply<<<Bn * Fn * NCHUNK, 32, 0, stream>>>(X, S, wsf, out);
}